// MoE_13426067767888
// MI455X (gfx1250) — compile-verified
//
#include <hip/hip_runtime.h>
#include <math.h>

// ---------------------------------------------------------------------------
// MoE (top-2 of 8 experts) + shared SwiGLU expert, fp32, for MI455X (gfx1250).
// All GEMMs use V_WMMA_F32_16X16X4_F32 (wave32 matrix pipe, full fp32).
// Tiling: 32 rows x 64 cols per 128-thread block; each wave owns two 16-row
// A fragments that share every B fragment (2x arithmetic intensity on B).
// ---------------------------------------------------------------------------

typedef float v2f __attribute__((ext_vector_type(2)));
typedef float v8f __attribute__((ext_vector_type(8)));

#define T_TOK 8192
#define H_DIM 1024
#define D_EXP 512
#define DS_DIM 1024
#define NEXP 8
#define TOPK 2
#define CAP 16640            // T*K + E*32 padding slots (32-aligned regions)
#define NTILES (CAP / 32)    // 520 32-row tiles
#define KC 256               // K-chunk staged in LDS (32*256*4 = 32 KB)

__device__ __forceinline__ v8f wmma4(v2f a, v2f b, v8f c) {
  // D = A(16x4 f32) * B(4x16 f32) + C(16x16 f32)
  return __builtin_amdgcn_wmma_f32_16x16x4_f32(false, a, false, b, (short)0, c,
                                               false, false);
}

__device__ __forceinline__ float silu_f(float g) {
  return g / (1.0f + expf(-g));
}

// ---------------------------------------------------------------------------
// Small bookkeeping kernels
// ---------------------------------------------------------------------------
__global__ void init_counts_kernel(int* counts) {
  if (threadIdx.x < NEXP) counts[threadIdx.x] = 0;
}

// Router: logits = x @ W_router, softmax, top-2, renormalize. Also per-block
// partial sums of softmax probs for the aux loss, and per-expert counts.
__global__ void __launch_bounds__(256)
router_kernel(const float* __restrict__ x, const float* __restrict__ Wr,
              int* __restrict__ eids, float* __restrict__ wts,
              int* __restrict__ counts, float* __restrict__ partial) {
  __shared__ float Wl[H_DIM * NEXP];  // 32 KB
  __shared__ float red[256];
  for (int i = threadIdx.x; i < H_DIM * NEXP; i += 256) Wl[i] = Wr[i];
  __syncthreads();

  const int t = blockIdx.x * 256 + threadIdx.x;
  const float* xr = x + (long)t * H_DIM;
  float lg[NEXP] = {0.f, 0.f, 0.f, 0.f, 0.f, 0.f, 0.f, 0.f};
  for (int h = 0; h < H_DIM; ++h) {
    float xv = xr[h];
#pragma unroll
    for (int e = 0; e < NEXP; ++e) lg[e] += xv * Wl[h * NEXP + e];
  }
  float m = lg[0];
#pragma unroll
  for (int e = 1; e < NEXP; ++e) m = fmaxf(m, lg[e]);
  float p[NEXP], s = 0.f;
#pragma unroll
  for (int e = 0; e < NEXP; ++e) { p[e] = expf(lg[e] - m); s += p[e]; }
  float inv = 1.0f / s;
#pragma unroll
  for (int e = 0; e < NEXP; ++e) p[e] *= inv;

  // top-2 (first occurrence on ties, matching lax.top_k)
  int i0 = 0;
#pragma unroll
  for (int e = 1; e < NEXP; ++e) if (p[e] > p[i0]) i0 = e;
  int i1 = -1; float b = -1.0f;
#pragma unroll
  for (int e = 0; e < NEXP; ++e)
    if (e != i0 && p[e] > b) { b = p[e]; i1 = e; }
  float w0 = p[i0], w1 = p[i1];
  float rs = 1.0f / (w0 + w1);
  w0 *= rs; w1 *= rs;

  eids[2 * t] = i0; eids[2 * t + 1] = i1;
  wts[2 * t] = w0;  wts[2 * t + 1] = w1;
  atomicAdd(&counts[i0], 1);
  atomicAdd(&counts[i1], 1);

  // block-reduce probs for aux loss
  for (int e = 0; e < NEXP; ++e) {
    red[threadIdx.x] = p[e];
    __syncthreads();
    for (int s2 = 128; s2 > 0; s2 >>= 1) {
      if (threadIdx.x < s2) red[threadIdx.x] += red[threadIdx.x + s2];
      __syncthreads();
    }
    if (threadIdx.x == 0) partial[blockIdx.x * NEXP + e] = red[0];
    __syncthreads();
  }
}

// Aux loss + padded-to-32 per-expert offsets + cursors (single thread; tiny).
__global__ void finalize_router_kernel(const float* __restrict__ partial,
                                       const int* __restrict__ counts,
                                       int* __restrict__ offsets,
                                       int* __restrict__ cursor,
                                       float* __restrict__ aux_out) {
  if (threadIdx.x == 0) {
    float aux = 0.f;
    for (int e = 0; e < NEXP; ++e) {
      float s = 0.f;
      for (int bk = 0; bk < T_TOK / 256; ++bk) s += partial[bk * NEXP + e];
      float mean = s / (float)T_TOK;
      aux += mean * mean;
    }
    aux_out[0] = (float)NEXP * aux;
    int off = 0;
    for (int e = 0; e < NEXP; ++e) {
      offsets[e] = off;
      cursor[e] = off;
      off += ((counts[e] + 31) >> 5) << 5;   // pad each region to 32 rows
    }
    offsets[NEXP] = off;
  }
}

__global__ void fill_perm_kernel(int* __restrict__ perm_tok,
                                 float* __restrict__ perm_w) {
  int i = blockIdx.x * 256 + threadIdx.x;
  if (i < CAP) { perm_tok[i] = 0; perm_w[i] = 0.0f; }
}

__global__ void scatter_kernel(const int* __restrict__ eids,
                               const float* __restrict__ wts,
                               int* __restrict__ cursor,
                               int* __restrict__ perm_tok,
                               float* __restrict__ perm_w) {
  int t = blockIdx.x * 256 + threadIdx.x;
#pragma unroll
  for (int k = 0; k < TOPK; ++k) {
    int e = eids[2 * t + k];
    int pos = atomicAdd(&cursor[e], 1);
    perm_tok[pos] = t;
    perm_w[pos] = wts[2 * t + k];
  }
}

__global__ void zero_f32_kernel(float* __restrict__ p) {
  p[(long)blockIdx.x * 1024 + threadIdx.x] = 0.0f;
}

__global__ void final_add_kernel(float* __restrict__ out,
                                 const float* __restrict__ routed) {
  long i = (long)blockIdx.x * 1024 + threadIdx.x;
  out[i] += routed[i];
}

// ---------------------------------------------------------------------------
// Shared expert gate+up: hs = silu(x@Wsg) * (x@Wsu)   [T, DS]
// grid (T/32, DS/64), 128 threads = 4 waves; wave -> one 16-col subtile,
// two 16-row A fragments sharing each B fragment.
// ---------------------------------------------------------------------------
__global__ void __launch_bounds__(128)
shared_gateup_kernel(const float* __restrict__ x, const float* __restrict__ Wsg,
                     const float* __restrict__ Wsu, float* __restrict__ hs) {
  __shared__ float Xl[32 * KC];  // 32 KB A-tile chunk
  const int mtile = blockIdx.x, ntile = blockIdx.y;
  const int wave = threadIdx.x >> 5, lane = threadIdx.x & 31;
  const int l = lane & 15, half = lane >> 4;
  const int col0 = ntile * 64 + wave * 16;
  const int row0 = mtile * 32;
  v8f accg0 = {0, 0, 0, 0, 0, 0, 0, 0}, accg1 = {0, 0, 0, 0, 0, 0, 0, 0};
  v8f accu0 = {0, 0, 0, 0, 0, 0, 0, 0}, accu1 = {0, 0, 0, 0, 0, 0, 0, 0};
  for (int k0 = 0; k0 < H_DIM; k0 += KC) {
    for (int idx = threadIdx.x; idx < 32 * KC; idx += 128) {
      int r = idx >> 8, c = idx & (KC - 1);
      Xl[idx] = x[(long)(row0 + r) * H_DIM + k0 + c];
    }
    __syncthreads();
    const float* Bg = Wsg + (long)(k0 + 2 * half) * DS_DIM + col0 + l;
    const float* Bu = Wsu + (long)(k0 + 2 * half) * DS_DIM + col0 + l;
    const float* A0 = Xl + l * KC + 2 * half;
    const float* A1 = Xl + (16 + l) * KC + 2 * half;
    for (int k = 0; k < KC; k += 4) {
      v2f a0; a0.x = A0[k]; a0.y = A0[k + 1];
      v2f a1; a1.x = A1[k]; a1.y = A1[k + 1];
      v2f bg; bg.x = Bg[(long)k * DS_DIM]; bg.y = Bg[(long)(k + 1) * DS_DIM];
      v2f bu; bu.x = Bu[(long)k * DS_DIM]; bu.y = Bu[(long)(k + 1) * DS_DIM];
      accg0 = wmma4(a0, bg, accg0);
      accg1 = wmma4(a1, bg, accg1);
      accu0 = wmma4(a0, bu, accu0);
      accu1 = wmma4(a1, bu, accu1);
    }
    __syncthreads();
  }
#pragma unroll
  for (int v = 0; v < 8; ++v) {
    int rl = v + 8 * half;
    hs[(long)(row0 + rl) * DS_DIM + col0 + l] = silu_f(accg0[v]) * accu0[v];
    hs[(long)(row0 + 16 + rl) * DS_DIM + col0 + l] = silu_f(accg1[v]) * accu1[v];
  }
}

// Shared expert down: out = hs @ Wsd   [T, H] -> writes d_out (shared part)
__global__ void __launch_bounds__(128)
shared_down_kernel(const float* __restrict__ hs, const float* __restrict__ Wsd,
                   float* __restrict__ out) {
  __shared__ float Al[32 * KC];
  const int mtile = blockIdx.x, ntile = blockIdx.y;
  const int wave = threadIdx.x >> 5, lane = threadIdx.x & 31;
  const int l = lane & 15, half = lane >> 4;
  const int col0 = ntile * 64 + wave * 16;
  const int row0 = mtile * 32;
  v8f acc0 = {0, 0, 0, 0, 0, 0, 0, 0}, acc1 = {0, 0, 0, 0, 0, 0, 0, 0};
  for (int k0 = 0; k0 < DS_DIM; k0 += KC) {
    for (int idx = threadIdx.x; idx < 32 * KC; idx += 128) {
      int r = idx >> 8, c = idx & (KC - 1);
      Al[idx] = hs[(long)(row0 + r) * DS_DIM + k0 + c];
    }
    __syncthreads();
    const float* Bp = Wsd + (long)(k0 + 2 * half) * H_DIM + col0 + l;
    const float* A0 = Al + l * KC + 2 * half;
    const float* A1 = Al + (16 + l) * KC + 2 * half;
    for (int k = 0; k < KC; k += 4) {
      v2f a0; a0.x = A0[k]; a0.y = A0[k + 1];
      v2f a1; a1.x = A1[k]; a1.y = A1[k + 1];
      v2f b; b.x = Bp[(long)k * H_DIM]; b.y = Bp[(long)(k + 1) * H_DIM];
      acc0 = wmma4(a0, b, acc0);
      acc1 = wmma4(a1, b, acc1);
    }
    __syncthreads();
  }
#pragma unroll
  for (int v = 0; v < 8; ++v) {
    int rl = v + 8 * half;
    out[(long)(row0 + rl) * H_DIM + col0 + l] = acc0[v];
    out[(long)(row0 + 16 + rl) * H_DIM + col0 + l] = acc1[v];
  }
}

// ---------------------------------------------------------------------------
// Grouped expert gate+up over the permuted token list (each 32-row tile is
// entirely one expert, guaranteed by 32-padding of expert regions).
// act[row, :] = silu(x[tok]@Wg[e]) * (x[tok]@Wu[e])   [CAP, D]
// ---------------------------------------------------------------------------
__global__ void __launch_bounds__(128)
expert_gateup_kernel(const float* __restrict__ x, const float* __restrict__ Wg,
                     const float* __restrict__ Wu,
                     const int* __restrict__ perm_tok,
                     const int* __restrict__ offsets,
                     float* __restrict__ act) {
  __shared__ float Xl[32 * KC];
  __shared__ int toks[32];
  __shared__ int e_sh;
  const int tile = blockIdx.x, row0 = tile * 32;
  if (threadIdx.x == 0) {
    int e = 0;
    for (int i = NEXP - 1; i >= 1; --i)
      if (row0 >= offsets[i]) { e = i; break; }
    e_sh = e;
  }
  if (threadIdx.x < 32) toks[threadIdx.x] = perm_tok[row0 + threadIdx.x];
  __syncthreads();

  const int wave = threadIdx.x >> 5, lane = threadIdx.x & 31;
  const int l = lane & 15, half = lane >> 4;
  const int col0 = blockIdx.y * 64 + wave * 16;
  const long wbase = (long)e_sh * H_DIM * D_EXP;
  v8f accg0 = {0, 0, 0, 0, 0, 0, 0, 0}, accg1 = {0, 0, 0, 0, 0, 0, 0, 0};
  v8f accu0 = {0, 0, 0, 0, 0, 0, 0, 0}, accu1 = {0, 0, 0, 0, 0, 0, 0, 0};
  for (int k0 = 0; k0 < H_DIM; k0 += KC) {
    for (int idx = threadIdx.x; idx < 32 * KC; idx += 128) {
      int r = idx >> 8, c = idx & (KC - 1);
      Xl[idx] = x[(long)toks[r] * H_DIM + k0 + c];
    }
    __syncthreads();
    const float* Bg = Wg + wbase + (long)(k0 + 2 * half) * D_EXP + col0 + l;
    const float* Bu = Wu + wbase + (long)(k0 + 2 * half) * D_EXP + col0 + l;
    const float* A0 = Xl + l * KC + 2 * half;
    const float* A1 = Xl + (16 + l) * KC + 2 * half;
    for (int k = 0; k < KC; k += 4) {
      v2f a0; a0.x = A0[k]; a0.y = A0[k + 1];
      v2f a1; a1.x = A1[k]; a1.y = A1[k + 1];
      v2f bg; bg.x = Bg[(long)k * D_EXP]; bg.y = Bg[(long)(k + 1) * D_EXP];
      v2f bu; bu.x = Bu[(long)k * D_EXP]; bu.y = Bu[(long)(k + 1) * D_EXP];
      accg0 = wmma4(a0, bg, accg0);
      accg1 = wmma4(a1, bg, accg1);
      accu0 = wmma4(a0, bu, accu0);
      accu1 = wmma4(a1, bu, accu1);
    }
    __syncthreads();
  }
#pragma unroll
  for (int v = 0; v < 8; ++v) {
    int rl = v + 8 * half;
    act[(long)(row0 + rl) * D_EXP + col0 + l] = silu_f(accg0[v]) * accu0[v];
    act[(long)(row0 + 16 + rl) * D_EXP + col0 + l] =
        silu_f(accg1[v]) * accu1[v];
  }
}

// Expert down-proj + weighted scatter: routed[tok] += w * (act_row @ Wd[e]).
// Exactly 2 nonzero contributions per routed element (+ exact +0.0 padding)
// -> bitwise deterministic despite atomics.
__global__ void __launch_bounds__(128)
expert_down_kernel(const float* __restrict__ act, const float* __restrict__ Wd,
                   const int* __restrict__ perm_tok,
                   const float* __restrict__ perm_w,
                   const int* __restrict__ offsets,
                   float* __restrict__ routed) {
  __shared__ float Al[32 * KC];
  __shared__ int toks[32];
  __shared__ float rowsw[32];
  __shared__ int e_sh;
  const int tile = blockIdx.x, row0 = tile * 32;
  if (threadIdx.x == 0) {
    int e = 0;
    for (int i = NEXP - 1; i >= 1; --i)
      if (row0 >= offsets[i]) { e = i; break; }
    e_sh = e;
  }
  if (threadIdx.x < 32) {
    toks[threadIdx.x] = perm_tok[row0 + threadIdx.x];
    rowsw[threadIdx.x] = perm_w[row0 + threadIdx.x];
  }
  __syncthreads();

  const int wave = threadIdx.x >> 5, lane = threadIdx.x & 31;
  const int l = lane & 15, half = lane >> 4;
  const int col0 = blockIdx.y * 64 + wave * 16;
  const long wbase = (long)e_sh * D_EXP * H_DIM;
  v8f acc0 = {0, 0, 0, 0, 0, 0, 0, 0}, acc1 = {0, 0, 0, 0, 0, 0, 0, 0};
  for (int k0 = 0; k0 < D_EXP; k0 += KC) {
    for (int idx = threadIdx.x; idx < 32 * KC; idx += 128) {
      int r = idx >> 8, c = idx & (KC - 1);
      Al[idx] = act[(long)(row0 + r) * D_EXP + k0 + c];
    }
    __syncthreads();
    const float* Bp = Wd + wbase + (long)(k0 + 2 * half) * H_DIM + col0 + l;
    const float* A0 = Al + l * KC + 2 * half;
    const float* A1 = Al + (16 + l) * KC + 2 * half;
    for (int k = 0; k < KC; k += 4) {
      v2f a0; a0.x = A0[k]; a0.y = A0[k + 1];
      v2f a1; a1.x = A1[k]; a1.y = A1[k + 1];
      v2f b; b.x = Bp[(long)k * H_DIM]; b.y = Bp[(long)(k + 1) * H_DIM];
      acc0 = wmma4(a0, b, acc0);
      acc1 = wmma4(a1, b, acc1);
    }
    __syncthreads();
  }
#pragma unroll
  for (int v = 0; v < 8; ++v) {
    int rl = v + 8 * half;
    atomicAdd(&routed[(long)toks[rl] * H_DIM + col0 + l],
              rowsw[rl] * acc0[v]);
    atomicAdd(&routed[(long)toks[rl + 16] * H_DIM + col0 + l],
              rowsw[rl + 16] * acc1[v]);
  }
}

// ---------------------------------------------------------------------------
// Host-side launch sequence (graph-capture safe: only kernel launches).
// Inputs: x, W_router, Wg, Wu, Wd, Wsg, Wsu, Wsd  (all fp32)
// Output: [T*H] out then [1] aux_loss
// ---------------------------------------------------------------------------
extern "C" void kernel_launch(void* const* d_in, const int* in_sizes, int n_in,
                              void* d_out, int out_size, void* d_ws,
                              size_t ws_size, hipStream_t stream) {
  const float* x   = (const float*)d_in[0];
  const float* Wr  = (const float*)d_in[1];
  const float* Wg  = (const float*)d_in[2];
  const float* Wu  = (const float*)d_in[3];
  const float* Wd  = (const float*)d_in[4];
  const float* Wsg = (const float*)d_in[5];
  const float* Wsu = (const float*)d_in[6];
  const float* Wsd = (const float*)d_in[7];
  float* out = (float*)d_out;

  // workspace layout
  char* ws = (char*)d_ws;
  float* hs_routed = (float*)ws;  // 32 MB: hs first, then reused as routed
  size_t off = (size_t)T_TOK * H_DIM * 4;
  float* act = (float*)(ws + off);
  off += (size_t)CAP * D_EXP * 4;
  int* eids = (int*)(ws + off);           off += (size_t)2 * T_TOK * 4;
  float* wts = (float*)(ws + off);        off += (size_t)2 * T_TOK * 4;
  int* perm_tok = (int*)(ws + off);       off += (size_t)CAP * 4;
  float* perm_w = (float*)(ws + off);     off += (size_t)CAP * 4;
  int* counts = (int*)(ws + off);         off += 64;   // 8 ints + pad
  int* cursor = (int*)(ws + off);         off += 64;   // 8 ints + pad
  int* offsets = (int*)(ws + off);        off += 64;   // 9 ints + pad
  float* partial = (float*)(ws + off);    // (T/256)*8 floats

  // 1) routing
  init_counts_kernel<<<1, 32, 0, stream>>>(counts);
  router_kernel<<<T_TOK / 256, 256, 0, stream>>>(x, Wr, eids, wts, counts,
                                                 partial);
  finalize_router_kernel<<<1, 32, 0, stream>>>(partial, counts, offsets,
                                               cursor, out + (size_t)T_TOK * H_DIM);
  fill_perm_kernel<<<(CAP + 255) / 256, 256, 0, stream>>>(perm_tok, perm_w);
  scatter_kernel<<<T_TOK / 256, 256, 0, stream>>>(eids, wts, cursor, perm_tok,
                                                  perm_w);

  // 2) shared expert (uses hs region)
  shared_gateup_kernel<<<dim3(T_TOK / 32, DS_DIM / 64), 128, 0, stream>>>(
      x, Wsg, Wsu, hs_routed);
  shared_down_kernel<<<dim3(T_TOK / 32, H_DIM / 64), 128, 0, stream>>>(
      hs_routed, Wsd, out);

  // 3) routed experts (hs region now reused as zeroed 'routed' accumulator)
  zero_f32_kernel<<<(T_TOK * H_DIM) / 1024, 1024, 0, stream>>>(hs_routed);
  expert_gateup_kernel<<<dim3(NTILES, D_EXP / 64), 128, 0, stream>>>(
      x, Wg, Wu, perm_tok, offsets, act);
  expert_down_kernel<<<dim3(NTILES, H_DIM / 64), 128, 0, stream>>>(
      act, Wd, perm_tok, perm_w, offsets, hs_routed);

  // 4) out = shared + routed
  final_add_kernel<<<(T_TOK * H_DIM) / 1024, 1024, 0, stream>>>(out, hs_routed);
}